// Fp16SegMla_67070209295098
// MI455X (gfx1250) — compile-verified
//
#include <hip/hip_runtime.h>

typedef float v2f __attribute__((ext_vector_type(2)));
typedef float v8f __attribute__((ext_vector_type(8)));
typedef unsigned int v4u __attribute__((ext_vector_type(4)));
typedef int v4i __attribute__((ext_vector_type(4)));
typedef int v8i __attribute__((ext_vector_type(8)));

#define B_SZ    64
#define NHEAD   16
#define MAX_LEN 4096
#define DKV     576
#define VD      512
#define NS      4        // sequence splits (flash-decoding)
#define SPLITL  1024
#define TS      16       // seq tile
#define WAVES   4
#define THREADS 128
#define SM_SCALE 0.041666666666666664f
#define NEG_LARGE (-3.0e38f)

// TDM writes the 16x576 f32 tile with 2 pad dwords every 32 dwords:
// LDS row stride = 576 + 36 = 612 dwords -> 16 seq/v rows hit 16 distinct banks,
// and padded offsets stay even -> b64 LDS loads remain legal.
#define KPAD    9792     // 16 * 612 dwords per buffer
#define SMEM_FLOATS (2 * KPAD + WAVES * 8 * 32 + WAVES * 16 * 18)
#define SMEM_BYTES  (SMEM_FLOATS * 4)

__device__ __forceinline__ unsigned padoff(unsigned g) { return g + ((g >> 5) << 1); }

// Issue one TDM load of a 16-row x 576-dword tile into LDS (padded).
// D# group0/group1 per CDNA5 ISA 8.3/8.4; groups 2/3 unused (2-D tensor).
__device__ __forceinline__ void tdm_load_tile(unsigned lds_addr, const float* gsrc) {
    unsigned long long ga = (unsigned long long)(uintptr_t)gsrc;
    v4u g0;
    g0[0] = 1u;                                                   // count=1
    g0[1] = (unsigned)__builtin_amdgcn_readfirstlane((int)lds_addr);
    g0[2] = (unsigned)__builtin_amdgcn_readfirstlane((int)(unsigned)(ga & 0xFFFFFFFFull));
    g0[3] = ((unsigned)__builtin_amdgcn_readfirstlane((int)(unsigned)((ga >> 32) & 0x1FFFFFFull)))
            | (2u << 30);                                         // type=2 (image)
    v8i g1;
    // data_size=4B, pad_enable, pad_interval=32 dwords (code 4), pad_amount=2 dwords (field 1)
    g1[0] = (int)((2u << 16) | (1u << 20) | (4u << 22) | (1u << 25));
    g1[1] = (int)((DKV & 0xFFFFu) << 16);                // tensor_dim0[15:0]
    g1[2] = (int)((DKV >> 16) | ((TS & 0xFFFFu) << 16)); // tensor_dim0[31:16] | tensor_dim1[15:0]
    g1[3] = (int)((unsigned)DKV << 16);                  // tensor_dim1[31:16]=0 | tile_dim0=576
    g1[4] = (int)TS;                                     // tile_dim1=16, tile_dim2=0
    g1[5] = (int)DKV;                                    // tensor_dim0_stride[31:0]=576
    g1[6] = 0;                                           // stride0 hi | stride1 lo
    g1[7] = 0;
    v4i z4 = {0, 0, 0, 0};
    v8i z8 = {0, 0, 0, 0, 0, 0, 0, 0};
    __builtin_amdgcn_tensor_load_to_lds(g0, g1, z4, z4, z8, 0);
}

// ---------------------------------------------------------------------------
// Kernel 1: per (batch, seq-split) flash attention over the compressed latent
// cache. fp32 WMMA 16x16x4; K/V tiles DMA'd by the Tensor Data Mover with
// double buffering (TENSORcnt pipelining).
// ---------------------------------------------------------------------------
__global__ __launch_bounds__(THREADS)
void mla_split_kernel(const float* __restrict__ q,      // [B,H,DKV]
                      const float* __restrict__ kvs,    // [B,1,DKV]
                      const float* __restrict__ cache,  // [B,MAX_LEN,DKV]
                      const int*   __restrict__ ctx,    // [B]
                      float* __restrict__ O_ws,         // [B,NS,H,VD]
                      float* __restrict__ m_ws,         // [B,NS,H]
                      float* __restrict__ l_ws)         // [B,NS,H]
{
    extern __shared__ float smem[];
    float* sK0   = smem;                         // padded K/V tile, buffer 0
    float* sK1   = smem + KPAD;                  // buffer 1
    float* sPart = smem + 2 * KPAD;              // partial scores [wave][i][lane]
    float* sP    = sPart + WAVES * 8 * 32;       // per-wave P tile [head][seq]

    const int b     = blockIdx.x / NS;
    const int split = blockIdx.x % NS;
    const int tid   = threadIdx.x;
    const int wave  = tid >> 5;
    const int lane  = tid & 31;
    const int ln16  = lane & 15;
    const int hi    = lane >> 4;                 // 0 or 1 (lane half)

    const int ctxlen = ctx[b];
    const int total  = ctxlen + 1;
    const int base   = split * SPLITL;
    int count = total - base;
    if (count > SPLITL) count = SPLITL;
    const int ntiles = (count > 0) ? ((count + TS - 1) / TS) : 0;

    const unsigned ldsK0 = (unsigned)(uintptr_t)(void*)sK0;
    const unsigned ldsK1 = (unsigned)(uintptr_t)(void*)sK1;
    const float* cbase = cache + (size_t)b * MAX_LEN * DKV;

    // --- Q A-fragments for this wave's 144-dim slice, kept in registers.
    // A 16x4 f32 layout: lane(0..15)=head M, VGPR0 holds K=d0+2*hi, VGPR1 K=+1.
    const int dbase = wave * (DKV / WAVES);
    v2f aq[36];
    {
        const float* qrow = q + ((size_t)b * NHEAD + ln16) * DKV;
        #pragma unroll
        for (int kk = 0; kk < 36; ++kk) {
            const float2 t2 = *(const float2*)(qrow + dbase + kk * 4 + hi * 2);
            aq[kk].x = t2.x; aq[kk].y = t2.y;
        }
    }

    // accumulators: O is 8 N-tiles of 16 v-dims (this wave owns v [wave*128,+128))
    v8f o[8];
    #pragma unroll
    for (int t = 0; t < 8; ++t)
        #pragma unroll
        for (int i = 0; i < 8; ++i) o[t][i] = 0.0f;
    float mrow[8], lrow[8];
    #pragma unroll
    for (int i = 0; i < 8; ++i) { mrow[i] = NEG_LARGE; lrow[i] = 0.0f; }

    // ---- prime the TDM pipeline: tile 0 -> buffer 0
    if (wave == 0 && ntiles > 0)
        tdm_load_tile(ldsK0, cbase + (size_t)base * DKV);

    for (int tile = 0; tile < ntiles; ++tile) {
        const int s0 = base + tile * TS;
        float* bufp = (tile & 1) ? sK1 : sK0;

        // ---- issue DMA for the next tile, then drain this tile's DMA
        if (wave == 0) {
            if (tile + 1 < ntiles) {
                tdm_load_tile((tile & 1) ? ldsK0 : ldsK1,
                              cbase + (size_t)(s0 + TS) * DKV);
                __builtin_amdgcn_s_wait_tensorcnt(1);
            } else {
                __builtin_amdgcn_s_wait_tensorcnt(0);
            }
        }
        __syncthreads();

        // ---- fold in the decode-step cache update: row ctxlen comes from kvs
        {
            const int prow = ctxlen - s0;
            if (prow >= 0 && prow < TS) {          // block-uniform branch
                for (int idx = tid; idx < DKV; idx += THREADS) {
                    unsigned g = (unsigned)(prow * DKV + idx);
                    bufp[padoff(g)] = kvs[(size_t)b * DKV + idx];
                }
            }
        }
        __syncthreads();

        // ---- scores: partial Q.K^T over this wave's 144 dims (36 WMMAs).
        // Issue loads in groups of 6 ahead of their WMMAs so LDS latency
        // overlaps the matrix pipe instead of s_wait_dscnt 0 per step.
        v8f c;
        #pragma unroll
        for (int i = 0; i < 8; ++i) c[i] = 0.0f;
        const unsigned gq = (unsigned)(ln16 * DKV);
        #pragma unroll
        for (int grp = 0; grp < 6; ++grp) {
            v2f bgrp[6];
            #pragma unroll
            for (int j = 0; j < 6; ++j) {
                const int kk = grp * 6 + j;
                const unsigned ga = gq + (unsigned)(dbase + kk * 4 + hi * 2);
                const float2 t2 = *(const float2*)(&bufp[padoff(ga)]);  // even offset -> b64
                bgrp[j].x = t2.x; bgrp[j].y = t2.y;
            }
            #pragma unroll
            for (int j = 0; j < 6; ++j) {
                c = __builtin_amdgcn_wmma_f32_16x16x4_f32(
                        false, aq[grp * 6 + j], false, bgrp[j], (short)0, c, false, false);
            }
        }
        #pragma unroll
        for (int i = 0; i < 8; ++i)
            sPart[(wave * 8 + i) * 32 + lane] = c[i];
        __syncthreads();

        // ---- full scores + mask + online softmax (redundant per wave)
        v8f s;
        const bool valid = (s0 + ln16) < total;
        #pragma unroll
        for (int i = 0; i < 8; ++i) {
            float acc = 0.0f;
            #pragma unroll
            for (int w = 0; w < WAVES; ++w)
                acc += sPart[(w * 8 + i) * 32 + lane];
            acc *= SM_SCALE;
            s[i] = valid ? acc : -1.0e30f;
        }
        float scl[8];
        #pragma unroll
        for (int i = 0; i < 8; ++i) {
            float rm = s[i];                     // 16-lane row max (per head)
            rm = fmaxf(rm, __shfl_xor(rm, 1, 32));
            rm = fmaxf(rm, __shfl_xor(rm, 2, 32));
            rm = fmaxf(rm, __shfl_xor(rm, 4, 32));
            rm = fmaxf(rm, __shfl_xor(rm, 8, 32));
            const float mnew = fmaxf(mrow[i], rm);
            const float sc   = __expf(mrow[i] - mnew);
            float p  = __expf(s[i] - mnew);      // masked lanes underflow to 0
            float rs = p;
            rs += __shfl_xor(rs, 1, 32);
            rs += __shfl_xor(rs, 2, 32);
            rs += __shfl_xor(rs, 4, 32);
            rs += __shfl_xor(rs, 8, 32);
            lrow[i] = lrow[i] * sc + rs;
            mrow[i] = mnew;
            scl[i]  = sc;
            s[i]    = p;
        }
        #pragma unroll
        for (int t = 0; t < 8; ++t)
            #pragma unroll
            for (int i = 0; i < 8; ++i) o[t][i] *= scl[i];

        // ---- transpose P (head<->lane) through this wave's private LDS patch
        {
            float* pw = &sP[wave * 16 * 18];
            #pragma unroll
            for (int i = 0; i < 8; ++i)
                pw[(i + hi * 8) * 18 + ln16] = s[i];
        }

        // ---- O += P.V over this wave's 128 v-dims (8 tiles x 4 K-steps).
        // For each K-step: batch all 16 V-fragment loads, then 8 WMMAs.
        {
            const float* pw = &sP[wave * 16 * 18];
            #pragma unroll
            for (int k0 = 0; k0 < 16; k0 += 4) {
                v2f ap;
                const float2 t2 = *(const float2*)(&pw[ln16 * 18 + k0 + hi * 2]);
                ap.x = t2.x; ap.y = t2.y;
                const unsigned grow0 = (unsigned)((k0 + hi * 2) * DKV);
                const unsigned grow1 = grow0 + DKV;
                v2f bvv[8];
                #pragma unroll
                for (int t = 0; t < 8; ++t) {
                    const unsigned v0 = (unsigned)(wave * 128 + t * 16 + ln16);
                    bvv[t].x = bufp[padoff(grow0 + v0)];
                    bvv[t].y = bufp[padoff(grow1 + v0)];
                }
                #pragma unroll
                for (int t = 0; t < 8; ++t) {
                    o[t] = __builtin_amdgcn_wmma_f32_16x16x4_f32(
                               false, ap, false, bvv[t], (short)0, o[t], false, false);
                }
            }
        }
        __syncthreads();   // all waves done with bufp/sPart before next DMA reuses it
    }

    // ---- write unnormalized partials
    float* Ob = O_ws + (((size_t)b * NS + split) * NHEAD) * VD;
    #pragma unroll
    for (int t = 0; t < 8; ++t) {
        const int v0 = wave * 128 + t * 16;
        #pragma unroll
        for (int i = 0; i < 8; ++i) {
            const int h = i + hi * 8;
            Ob[(size_t)h * VD + v0 + ln16] = o[t][i];
        }
    }
    if (wave == 0 && ln16 == 0) {      // lanes 0 and 16 cover heads 0..7 / 8..15
        #pragma unroll
        for (int i = 0; i < 8; ++i) {
            const int h = i + hi * 8;
            m_ws[((size_t)b * NS + split) * NHEAD + h] = mrow[i];
            l_ws[((size_t)b * NS + split) * NHEAD + h] = lrow[i];
        }
    }
}

// ---------------------------------------------------------------------------
// Kernel 2: merge the NS split partials with the standard log-sum-exp combine.
// ---------------------------------------------------------------------------
__global__ __launch_bounds__(128)
void mla_combine_kernel(const float* __restrict__ O_ws,
                        const float* __restrict__ m_ws,
                        const float* __restrict__ l_ws,
                        float* __restrict__ out)        // [B,H,VD]
{
    const int b = blockIdx.x;
    const int h = blockIdx.y;
    const int tid = threadIdx.x;

    float mi[NS], wi[NS];
    float M = NEG_LARGE;
    #pragma unroll
    for (int i = 0; i < NS; ++i) {
        mi[i] = m_ws[((size_t)b * NS + i) * NHEAD + h];
        M = fmaxf(M, mi[i]);
    }
    float denom = 0.0f;
    #pragma unroll
    for (int i = 0; i < NS; ++i) {
        wi[i] = __expf(mi[i] - M);
        denom += wi[i] * l_ws[((size_t)b * NS + i) * NHEAD + h];
    }
    const float inv = 1.0f / denom;

    for (int v = tid; v < VD; v += 128) {
        float acc = 0.0f;
        #pragma unroll
        for (int i = 0; i < NS; ++i)
            acc += wi[i] * O_ws[(((size_t)b * NS + i) * NHEAD + h) * (size_t)VD + v];
        out[((size_t)b * NHEAD + h) * VD + v] = acc * inv;
    }
}

extern "C" void kernel_launch(void* const* d_in, const int* in_sizes, int n_in,
                              void* d_out, int out_size, void* d_ws, size_t ws_size,
                              hipStream_t stream) {
    const float* q     = (const float*)d_in[0];   // [B,H,576]
    const float* kvs   = (const float*)d_in[1];   // [B,1,576]
    const float* cache = (const float*)d_in[2];   // [B,4096,576]
    const int*   ctx   = (const int*)d_in[3];     // [B]
    float* out = (float*)d_out;

    float* O_ws = (float*)d_ws;                               // B*NS*H*VD floats
    float* m_ws = O_ws + (size_t)B_SZ * NS * NHEAD * VD;      // B*NS*H
    float* l_ws = m_ws + (size_t)B_SZ * NS * NHEAD;           // B*NS*H

    (void)hipFuncSetAttribute((const void*)mla_split_kernel,
                              hipFuncAttributeMaxDynamicSharedMemorySize,
                              SMEM_BYTES);

    mla_split_kernel<<<dim3(B_SZ * NS), dim3(THREADS), SMEM_BYTES, stream>>>(
        q, kvs, cache, ctx, O_ws, m_ws, l_ws);
    mla_combine_kernel<<<dim3(B_SZ, NHEAD), dim3(128), 0, stream>>>(
        O_ws, m_ws, l_ws, out);
}